// GraphClassifier_48473000903347
// MI455X (gfx1250) — compile-verified
//
#include <hip/hip_runtime.h>
#include <hip/hip_bf16.h>

#define N_NODES   100000
#define N_EDGES   1600000
#define N_GRAPHS  64
#define D_IN      3
#define HIDDEN    256
#define N_CLASSES 10
#define NEG_SLOPE 0.01f

typedef __attribute__((ext_vector_type(2))) float v2f;
typedef __attribute__((ext_vector_type(8))) float v8f;

// ---------------------------------------------------------------------------
// Degree: deg[i] = 1 + indegree(i)   (self loop counted)
// ---------------------------------------------------------------------------
__global__ void init_deg_kernel(float* deg, int n) {
    int i = blockIdx.x * blockDim.x + threadIdx.x;
    if (i < n) deg[i] = 1.0f;
}

__global__ void edge_deg_kernel(const int* __restrict__ dst, float* deg, int e) {
    int i = blockIdx.x * blockDim.x + threadIdx.x;
    if (i < e) atomicAdd(&deg[dst[i]], 1.0f);
}

__global__ void dinv_kernel(const float* __restrict__ deg, float* dinv, int n) {
    int i = blockIdx.x * blockDim.x + threadIdx.x;
    if (i < n) dinv[i] = rsqrtf(deg[i]);
}

// ---------------------------------------------------------------------------
// Layer 1 transform: h = x @ W1   (K = 3, scalar FMA is optimal here)
// ---------------------------------------------------------------------------
__global__ void xw1_kernel(const float* __restrict__ x, const float* __restrict__ W1,
                           float* __restrict__ h, int n) {
    int i = blockIdx.x * blockDim.x + threadIdx.x;
    if (i >= n * HIDDEN) return;
    int node = i >> 8;
    int f    = i & (HIDDEN - 1);
    float x0 = x[node * 3 + 0];
    float x1 = x[node * 3 + 1];
    float x2 = x[node * 3 + 2];
    h[i] = fmaf(x0, W1[f], fmaf(x1, W1[HIDDEN + f], x2 * W1[2 * HIDDEN + f]));
}

// ---------------------------------------------------------------------------
// WMMA fp32 GEMM:  C[M,256] = A[M,256] @ W[256,256]
// One wave computes a 16x64 output strip (4 x 16x16 tiles): the A fragment is
// loaded once per k-step and reused by 4 V_WMMA_F32_16X16X4_F32 ops.
// Block = 128 threads (4 waves); wave w covers columns [w*64, w*64+64).
// Grid = M/16 blocks.
// Fragment layouts per CDNA5 ISA 7.12.2:
//   A (16x4 f32): VGPR0 K={k, k+2} on lane halves, VGPR1 K={k+1, k+3};
//                 i.e. lanes 0-15 (M=lane) hold K={k,k+1}, lanes 16-31 K={k+2,k+3}
//   B (4x16 f32): VGPR0 rows {k, k+2} on lane halves, VGPR1 rows {k+1, k+3}
//   C (16x16 f32): VGPR r -> row r + 8*(lane>=16), col = lane&15
// ---------------------------------------------------------------------------
__global__ __launch_bounds__(128) void gemm_wmma_f32_kernel(
    const float* __restrict__ A, const float* __restrict__ W,
    float* __restrict__ C, int M) {
    const int wave = threadIdx.x >> 5;        // 0..3
    const int lane = threadIdx.x & 31;
    const int half = lane >> 4;               // 0: lanes 0-15, 1: lanes 16-31
    const int l16  = lane & 15;
    const int m0   = blockIdx.x * 16;
    const int n0   = wave * 64;

    v8f acc[4] = {};
    const float* arow = A + (size_t)(m0 + l16) * HIDDEN;

    for (int k0 = 0; k0 < HIDDEN; k0 += 4) {
        const int ka = k0 + 2 * half;
        v2f a = *(const v2f*)(arow + ka);                  // K = ka, ka+1
        const float* wp = W + (size_t)ka * HIDDEN + n0 + l16;
#pragma unroll
        for (int t = 0; t < 4; ++t) {
            v2f b;
            b.x = wp[t * 16];                              // B row ka,   col n0+16t+l16
            b.y = wp[t * 16 + HIDDEN];                     // B row ka+1, same col
            acc[t] = __builtin_amdgcn_wmma_f32_16x16x4_f32(
                /*neg_a=*/false, a, /*neg_b=*/false, b,
                /*c_mod=*/(short)0, acc[t], /*reuse_a=*/false, /*reuse_b=*/false);
        }
    }

#pragma unroll
    for (int t = 0; t < 4; ++t) {
        float* crow = C + (size_t)(m0 + 8 * half) * HIDDEN + (n0 + t * 16 + l16);
#pragma unroll
        for (int r = 0; r < 8; ++r) {
            crow[(size_t)r * HIDDEN] = acc[t][r];
        }
    }
}

// ---------------------------------------------------------------------------
// Self-loop + bias pre-init: agg = h/deg + bias  (atomic accumulation target)
// ---------------------------------------------------------------------------
__global__ void node_init_kernel(const float* __restrict__ h, float* __restrict__ agg,
                                 const float* __restrict__ deg,
                                 const float* __restrict__ bias, int n) {
    int i = blockIdx.x * blockDim.x + threadIdx.x;
    if (i >= n * HIDDEN) return;
    int node = i >> 8;
    int f    = i & (HIDDEN - 1);
    agg[i] = h[i] / deg[node] + bias[f];
}

// ---------------------------------------------------------------------------
// Edge aggregation: agg[dst] += h[src] * dinv[src]*dinv[dst]
// One wave per edge, 32 lanes x 8 features; atomics land in L2 (102 MB < 192 MB).
// ---------------------------------------------------------------------------
__global__ __launch_bounds__(256) void edge_agg_kernel(
    const float* __restrict__ h, const int* __restrict__ src,
    const int* __restrict__ dst, const float* __restrict__ dinv,
    float* __restrict__ agg, int e) {
    int edge = blockIdx.x * 8 + (threadIdx.x >> 5);
    if (edge >= e) return;
    int lane = threadIdx.x & 31;
    int s = src[edge];
    int d = dst[edge];
    float coef = dinv[s] * dinv[d];
    const float* hs = h + (size_t)s * HIDDEN;
    float* ad = agg + (size_t)d * HIDDEN;
#pragma unroll
    for (int j = 0; j < 8; ++j) {
        int f = lane + 32 * j;
        atomicAdd(&ad[f], hs[f] * coef);
    }
}

// ---------------------------------------------------------------------------
// In-place leaky ReLU
// ---------------------------------------------------------------------------
__global__ void leaky_kernel(float* __restrict__ v, int n) {
    int i = blockIdx.x * blockDim.x + threadIdx.x;
    if (i >= n) return;
    float x = v[i];
    v[i] = (x >= 0.0f) ? x : NEG_SLOPE * x;
}

// ---------------------------------------------------------------------------
// Global mean pool (sums + counts via atomics into tiny L2-resident arrays)
// ---------------------------------------------------------------------------
__global__ void pool_kernel(const float* __restrict__ h, const int* __restrict__ batch,
                            float* __restrict__ pooled, int n) {
    int i = blockIdx.x * blockDim.x + threadIdx.x;
    if (i >= n * HIDDEN) return;
    int node = i >> 8;
    int f    = i & (HIDDEN - 1);
    atomicAdd(&pooled[(size_t)batch[node] * HIDDEN + f], h[i]);
}

__global__ void count_kernel(const int* __restrict__ batch, float* __restrict__ cnt, int n) {
    int i = blockIdx.x * blockDim.x + threadIdx.x;
    if (i < n) atomicAdd(&cnt[batch[i]], 1.0f);
}

// ---------------------------------------------------------------------------
// Classifier: logits = (pooled/cnt) @ Wc + bc, then softmax. 64 rows -> 64 threads.
// ---------------------------------------------------------------------------
__global__ void classify_kernel(const float* __restrict__ pooled, const float* __restrict__ cnt,
                                const float* __restrict__ Wc, const float* __restrict__ bc,
                                float* __restrict__ out) {
    int g = blockIdx.x * blockDim.x + threadIdx.x;
    if (g >= N_GRAPHS) return;
    float inv = 1.0f / fmaxf(cnt[g], 1.0f);
    float logits[N_CLASSES];
#pragma unroll
    for (int c = 0; c < N_CLASSES; ++c) logits[c] = bc[c];
    const float* p = pooled + (size_t)g * HIDDEN;
    for (int k = 0; k < HIDDEN; ++k) {
        float pk = p[k] * inv;
#pragma unroll
        for (int c = 0; c < N_CLASSES; ++c)
            logits[c] = fmaf(pk, Wc[k * N_CLASSES + c], logits[c]);
    }
    float mx = logits[0];
#pragma unroll
    for (int c = 1; c < N_CLASSES; ++c) mx = fmaxf(mx, logits[c]);
    float sum = 0.0f;
#pragma unroll
    for (int c = 0; c < N_CLASSES; ++c) { logits[c] = __expf(logits[c] - mx); sum += logits[c]; }
    float rs = 1.0f / sum;
#pragma unroll
    for (int c = 0; c < N_CLASSES; ++c) out[(size_t)g * N_CLASSES + c] = logits[c] * rs;
}

// ---------------------------------------------------------------------------
// Launch
// ---------------------------------------------------------------------------
extern "C" void kernel_launch(void* const* d_in, const int* in_sizes, int n_in,
                              void* d_out, int out_size, void* d_ws, size_t ws_size,
                              hipStream_t stream) {
    const float* x     = (const float*)d_in[0];
    const int*   ei    = (const int*)d_in[1];       // [2, E]
    const int*   batch = (const int*)d_in[2];
    const float* W1    = (const float*)d_in[3];
    const float* b1    = (const float*)d_in[4];
    const float* W2    = (const float*)d_in[5];
    const float* b2    = (const float*)d_in[6];
    const float* W3    = (const float*)d_in[7];
    const float* b3    = (const float*)d_in[8];
    const float* Wc    = (const float*)d_in[9];
    const float* bc    = (const float*)d_in[10];
    float* out = (float*)d_out;

    const int* src = ei;
    const int* dst = ei + N_EDGES;

    // Workspace carve-up (256B aligned)
    char* ws = (char*)d_ws;
    size_t off = 0;
    auto alloc = [&](size_t bytes) { char* p = ws + off; off = (off + bytes + 255) & ~(size_t)255; return p; };
    float* deg    = (float*)alloc((size_t)N_NODES * 4);
    float* dinv   = (float*)alloc((size_t)N_NODES * 4);
    float* bufH   = (float*)alloc((size_t)N_NODES * HIDDEN * 4);   // GEMM / transform output
    float* bufAgg = (float*)alloc((size_t)N_NODES * HIDDEN * 4);   // atomic target -> activations
    float* pooled = (float*)alloc((size_t)N_GRAPHS * HIDDEN * 4);
    float* cnt    = (float*)alloc((size_t)N_GRAPHS * 4);
    (void)ws_size; (void)n_in; (void)in_sizes; (void)out_size;

    const int NF = N_NODES * HIDDEN;           // 25.6M
    dim3 blk(256);
    dim3 grdNodes((N_NODES + 255) / 256);
    dim3 grdNF((NF + 255) / 256);
    dim3 grdEdges((N_EDGES + 255) / 256);
    dim3 grdEdgeAgg((N_EDGES + 7) / 8);        // 8 edges (waves) per block
    dim3 grdGemm(N_NODES / 16);                // 6250 blocks; 4 waves x 16x64 strip
    dim3 blkGemm(128);

    // Degrees (computed fresh each call: deterministic)
    init_deg_kernel<<<grdNodes, blk, 0, stream>>>(deg, N_NODES);
    edge_deg_kernel<<<grdEdges, blk, 0, stream>>>(dst, deg, N_EDGES);
    dinv_kernel<<<grdNodes, blk, 0, stream>>>(deg, dinv, N_NODES);

    // ---- Layer 1: h = x@W1 ; agg = h/deg + b1 ; += edges ; leaky -> bufAgg
    xw1_kernel<<<grdNF, blk, 0, stream>>>(x, W1, bufH, N_NODES);
    node_init_kernel<<<grdNF, blk, 0, stream>>>(bufH, bufAgg, deg, b1, N_NODES);
    edge_agg_kernel<<<grdEdgeAgg, blk, 0, stream>>>(bufH, src, dst, dinv, bufAgg, N_EDGES);
    leaky_kernel<<<grdNF, blk, 0, stream>>>(bufAgg, NF);

    // ---- Layer 2: h2 = act1@W2 (WMMA) ; agg ; leaky -> bufAgg
    gemm_wmma_f32_kernel<<<grdGemm, blkGemm, 0, stream>>>(bufAgg, W2, bufH, N_NODES);
    node_init_kernel<<<grdNF, blk, 0, stream>>>(bufH, bufAgg, deg, b2, N_NODES);
    edge_agg_kernel<<<grdEdgeAgg, blk, 0, stream>>>(bufH, src, dst, dinv, bufAgg, N_EDGES);
    leaky_kernel<<<grdNF, blk, 0, stream>>>(bufAgg, NF);

    // ---- Layer 3: h3 = act2@W3 (WMMA) ; agg ; leaky -> bufAgg
    gemm_wmma_f32_kernel<<<grdGemm, blkGemm, 0, stream>>>(bufAgg, W3, bufH, N_NODES);
    node_init_kernel<<<grdNF, blk, 0, stream>>>(bufH, bufAgg, deg, b3, N_NODES);
    edge_agg_kernel<<<grdEdgeAgg, blk, 0, stream>>>(bufH, src, dst, dinv, bufAgg, N_EDGES);
    leaky_kernel<<<grdNF, blk, 0, stream>>>(bufAgg, NF);

    // ---- Pool + classify
    hipMemsetAsync(pooled, 0, (size_t)N_GRAPHS * HIDDEN * 4, stream);
    hipMemsetAsync(cnt, 0, (size_t)N_GRAPHS * 4, stream);
    pool_kernel<<<grdNF, blk, 0, stream>>>(bufAgg, batch, pooled, N_NODES);
    count_kernel<<<grdNodes, blk, 0, stream>>>(batch, cnt, N_NODES);
    classify_kernel<<<1, 64, 0, stream>>>(pooled, cnt, Wc, bc, out);
}